// OptimalTransportLossFunc_69509750719068
// MI455X (gfx1250) — compile-verified
//
#include <hip/hip_runtime.h>
#include <hip/hip_bf16.h>
#include <math.h>

#define BB     16
#define NN     1024
#define NRR    1025
#define VCH    32
#define CHROWS 33          // 32 chunks * 33 >= 1025 rows
#define HBINS  4096
#define NITERS 50
#define LOGB   (-6.9314718055994531f)   // log(1/1024)
#define LOGAW  (-6.9314718055994531f)   // log(1/1024)

typedef float v2f __attribute__((ext_vector_type(2)));
typedef float v8f __attribute__((ext_vector_type(8)));

// ---- helpers -------------------------------------------------------------

// order-preserving float -> uint encoding (for atomic min/max)
__device__ __forceinline__ unsigned int fenc(float x) {
  unsigned int u = __float_as_uint(x);
  return (u & 0x80000000u) ? ~u : (u | 0x80000000u);
}
__device__ __forceinline__ float fdec(unsigned int e) {
  unsigned int u = (e & 0x80000000u) ? (e & 0x7FFFFFFFu) : ~e;
  return __uint_as_float(u);
}

// online (streaming) logsumexp accumulate: state (m,s) += value x
__device__ __forceinline__ void lse_acc(float& m, float& s, float x) {
  if (x == -INFINITY) return;                 // exp(-inf) contributes 0
  if (x > m) { s = s * __expf(m - x) + 1.0f; m = x; }
  else       { s += __expf(x - m); }
}
// merge two (m,s) logsumexp states
__device__ __forceinline__ void lse_merge(float& m, float& s, float m2, float s2) {
  if (s2 > 0.0f) {
    if (m2 > m) { s = s * __expf(m - m2) + s2; m = m2; }
    else        { s += s2 * __expf(m2 - m); }
  }
}

// ---- kernel 1: preprocessing --------------------------------------------
__global__ void ot_prep(const float* __restrict__ s_logits,
                        const unsigned char* __restrict__ mask_gt,
                        const float* __restrict__ xyz,
                        const float* __restrict__ xyz_gt,
                        const float* __restrict__ nph,
                        const float* __restrict__ nph_gt,
                        const unsigned char* __restrict__ valid,
                        const float* __restrict__ log_sigma,
                        const float* __restrict__ rnxyz,
                        const float* __restrict__ rnph,
                        float* __restrict__ gsA, float* __restrict__ xsB,
                        float* __restrict__ ri, float* __restrict__ cj,
                        float* __restrict__ loss, float* __restrict__ loga,
                        float* __restrict__ validrow,
                        float* __restrict__ u, float* __restrict__ v,
                        unsigned int* __restrict__ hist,
                        unsigned int* __restrict__ scal)
{
  int b = blockIdx.x, tid = threadIdx.x;
  float ls0 = log_sigma[0], ls1 = log_sigma[1], ls2 = log_sigma[2], ls3 = log_sigma[3];
  float sumlog = ls0 + ls1 + ls2 + ls3;
  float w0 = rnxyz[0] * __expf(-ls0);
  float w1 = rnxyz[1] * __expf(-ls1);
  float w2 = rnxyz[2] * __expf(-ls2);
  float w3 = rnph[0]  * __expf(-ls3);

  __shared__ int scnt[256];
  int cnt = 0;
  for (int j = tid; j < NN; j += 256) {
    int idx = b * NN + j;
    // prediction side (columns of cost)
    float x0 = xyz[idx*3+0]*w0, x1 = xyz[idx*3+1]*w1, x2 = xyz[idx*3+2]*w2;
    float x3 = nph[idx]*w3;
    *(float4*)(xsB + (unsigned)idx*4u) = make_float4(x0, x1, x2, x3);
    float s  = s_logits[idx];
    float t  = log1pf(__expf(-fabsf(s)));
    float lp1 = fmaxf(-s, 0.0f) + t;     // softplus(-s)
    float lp0 = fmaxf( s, 0.0f) + t;     // softplus(s)
    cj[idx] = 0.5f*(x0*x0 + x1*x1 + x2*x2 + x3*x3) + sumlog + lp1;
    loss[(unsigned)(b*NRR + NN)*NN + j] = lp0;    // appended lp0 row (i == N)
    // ground-truth side (rows of cost)
    float g0 = xyz_gt[idx*3+0]*w0, g1 = xyz_gt[idx*3+1]*w1, g2 = xyz_gt[idx*3+2]*w2;
    float g3 = nph_gt[idx]*w3;
    *(float4*)(gsA + (unsigned)idx*4u) = make_float4(g0, g1, g2, g3);
    ri[idx] = 0.5f*(g0*g0 + g1*g1 + g2*g2 + g3*g3);
    int mk = mask_gt[idx] ? 1 : 0;
    cnt += mk;
    loga[b*NRR + j] = mk ? LOGAW : -INFINITY;
    validrow[idx] = valid[idx] ? 1.0f : 0.0f;
    u[b*NRR + j] = 0.0f;
    v[idx] = 0.0f;
  }
  scnt[tid] = cnt; __syncthreads();
  for (int o = 128; o > 0; o >>= 1) { if (tid < o) scnt[tid] += scnt[tid+o]; __syncthreads(); }
  if (tid == 0) {
    float bg = 1.0f - (float)scnt[0] * (1.0f/(float)NN);
    loga[b*NRR + NN] = __logf(bg);
    u[b*NRR + NN] = 0.0f;
  }
  if (b == 0) {
    for (int k = tid; k < HBINS; k += 256) hist[k] = 0u;
    if (tid == 0) { scal[0] = 0xFFFFFFFFu; scal[1] = 0u; }
  }
}

// ---- kernel 2: cost matrix via WMMA f32 16x16x4 -------------------------
// One wave computes a 16-row stripe x 4 column-tiles: the A operand (16x4
// gt block) and the 8 ri values are loaded once and reused across 4
// back-to-back v_wmma_f32_16x16x4_f32 ops. 32-bit element offsets keep the
// backend on the saddr+voffset global addressing form.
__global__ void ot_cost_wmma(const float* __restrict__ gsA,
                             const float* __restrict__ xsB,
                             const float* __restrict__ ri,
                             const float* __restrict__ cj,
                             float* __restrict__ loss,
                             unsigned int* __restrict__ scal)
{
  int tid  = threadIdx.x;
  int wave = tid >> 5, lane = tid & 31;
  int wid  = blockIdx.x * 8 + wave;     // BB*64*16 = 16384 waves
  int b    = wid >> 10;                 // 64 ti * 16 tile-groups
  int ti   = (wid >> 4) & 63;
  int tj0  = (wid & 15) * 4;            // 4 consecutive column tiles
  int half = lane >> 4, lr = lane & 15;
  int kb   = half * 2;                  // A/B: lanes 0-15 K={0,1}, 16-31 K={2,3}

  unsigned abase = (unsigned)(b*NN + ti*16 + lr) * 4u + (unsigned)kb;
  v2f av = *(const v2f*)(gsA + abase);

  float riv[8];
  #pragma unroll
  for (int r = 0; r < 8; ++r) riv[r] = ri[(unsigned)(b*NN + ti*16 + r + half*8)];

  float lmin = INFINITY, lmax = -INFINITY;
  #pragma unroll
  for (int t = 0; t < 4; ++t) {
    int jcol = (tj0 + t)*16 + lr;
    v2f bv = *(const v2f*)(xsB + (unsigned)(b*NN + jcol)*4u + (unsigned)kb);
    v8f c = {};
    c = __builtin_amdgcn_wmma_f32_16x16x4_f32(false, av, false, bv,
                                              (short)0, c, false, false);
    float cjv = cj[(unsigned)(b*NN + jcol)];
    #pragma unroll
    for (int r = 0; r < 8; ++r) {
      int i = ti*16 + r + half*8;       // C/D layout: VGPR r -> M=r (+8 upper half)
      float val = riv[r] + cjv - c[r];
      loss[(unsigned)(b*NRR + i)*NN + (unsigned)jcol] = val;
      lmin = fminf(lmin, val); lmax = fmaxf(lmax, val);
    }
  }
  __shared__ float smn[256], smx[256];
  smn[tid] = lmin; smx[tid] = lmax; __syncthreads();
  for (int o = 128; o > 0; o >>= 1) {
    if (tid < o) { smn[tid] = fminf(smn[tid], smn[tid+o]);
                   smx[tid] = fmaxf(smx[tid], smx[tid+o]); }
    __syncthreads();
  }
  if (tid == 0) {
    atomicMin(&scal[0], fenc(smn[0]));
    atomicMax(&scal[1], fenc(smx[0]));
  }
}

// ---- kernel 3: decode min/max, set big ----------------------------------
__global__ void ot_range_init(unsigned int* __restrict__ scal)
{
  float* sf = (float*)scal;
  float lo = fdec(scal[0]), hi = fdec(scal[1]);
  sf[2] = lo; sf[3] = hi; sf[4] = 10.0f * hi;   // big = 10*max(loss)
}

// ---- kernel 4: histogram over the i<N rows (LDS-privatized, b128) -------
__global__ void ot_hist(const float* __restrict__ loss,
                        const unsigned int* __restrict__ scal,
                        unsigned int* __restrict__ hist)
{
  const float* sf = (const float*)scal;
  float lo = sf[2], hi = sf[3];
  float rng = hi - lo;
  float inv = (rng > 0.0f) ? (float)HBINS / rng : 0.0f;
  __shared__ unsigned int h[HBINS];
  for (int k = threadIdx.x; k < HBINS; k += 256) h[k] = 0u;
  __syncthreads();
  int stride = gridDim.x * 256;
  const int TOT4 = (BB*NN*NN) / 4;
  for (int t = blockIdx.x*256 + threadIdx.x; t < TOT4; t += stride) {
    int e = t * 4;
    int b = e >> 20;                    // NN*NN = 2^20
    int r = e & 0xFFFFF;
    float4 x = *(const float4*)(loss + (unsigned)b * (unsigned)(NRR*NN) + (unsigned)r);
    float xv[4] = {x.x, x.y, x.z, x.w};
    #pragma unroll
    for (int k = 0; k < 4; ++k) {
      int bin = (int)((xv[k] - lo) * inv);
      bin = bin < 0 ? 0 : (bin >= HBINS ? HBINS-1 : bin);
      atomicAdd(&h[bin], 1u);
    }
  }
  __syncthreads();
  for (int k = threadIdx.x; k < HBINS; k += 256) if (h[k]) atomicAdd(&hist[k], h[k]);
}

// ---- kernel 5: histogram select (two-level median) ----------------------
__global__ void ot_select(unsigned int* __restrict__ hist,
                          unsigned int* __restrict__ scal, int phase)
{
  float* sf = (float*)scal;
  if (threadIdx.x == 0) {
    float lo = sf[2], hi = sf[3];
    float wbin = (hi - lo) * (1.0f / (float)HBINS);
    unsigned long long tgt = (phase == 0)
        ? (unsigned long long)((long long)BB * NN * NN / 2)
        : (unsigned long long)scal[7];
    if (tgt == 0ull) tgt = 1ull;
    unsigned long long cum = 0ull, cbefore = 0ull;
    int bin = HBINS - 1;
    unsigned int cbin = 0u;
    for (int k = 0; k < HBINS; ++k) {
      unsigned int hk = hist[k];
      if (cum + hk >= tgt) { bin = k; cbin = hk; cbefore = cum; break; }
      cum += hk;
    }
    if (phase == 0) {
      sf[2] = lo + wbin * (float)bin;
      sf[3] = lo + wbin * (float)(bin + 1);
      scal[7] = (unsigned int)(tgt - cbefore);
    } else {
      float frac = (cbin > 0u) ? ((float)(tgt - cbefore) - 0.5f) / (float)cbin : 0.5f;
      float med  = lo + wbin * ((float)bin + frac);
      float reg  = 0.05f * med;
      float scale = -1.0f / reg;        // M = cost * scale
      sf[5] = scale;
      sf[6] = sf[4] * scale;            // bigM = -big/reg
    }
  }
  __syncthreads();
  if (phase == 0)
    for (int k = threadIdx.x; k < HBINS; k += blockDim.x) hist[k] = 0u;
}

// ---- kernel 6: v-update stage A: column LSE partials over row chunks ----
// Each thread owns 4 columns (float4 row loads); 32 row-chunks of 33 rows.
__global__ void ot_v_partial(const float* __restrict__ loss,
                             const float* __restrict__ u,
                             const float* __restrict__ validrow,
                             const unsigned int* __restrict__ scal,
                             float* __restrict__ vpm, float* __restrict__ vps)
{
  const float* sf = (const float*)scal;
  float scale = sf[5], bigM = sf[6];
  int blk = blockIdx.x;                 // BB * VCH = 512
  int b  = blk >> 5;
  int ch = blk & 31;
  int j0 = threadIdx.x * 4;
  int r0 = ch * CHROWS;
  int r1 = r0 + CHROWS; if (r1 > NRR) r1 = NRR;
  float m0 = -INFINITY, m1 = -INFINITY, m2 = -INFINITY, m3 = -INFINITY;
  float s0 = 0.0f, s1 = 0.0f, s2 = 0.0f, s3 = 0.0f;
  for (int i = r0; i < r1; ++i) {
    float ui = u[b*NRR + i];
    unsigned rowoff = (unsigned)(b*NRR + i) * NN;
    if (i == NN || validrow[b*NN + i] != 0.0f) {   // uniform per block
      __builtin_prefetch(loss + rowoff + NN + (unsigned)j0, 0, 0); // next row
      float4 lv = *(const float4*)(loss + rowoff + (unsigned)j0);
      lse_acc(m0, s0, scale*lv.x + ui);
      lse_acc(m1, s1, scale*lv.y + ui);
      lse_acc(m2, s2, scale*lv.z + ui);
      lse_acc(m3, s3, scale*lv.w + ui);
    } else {                                        // invalid row: skip load
      float x = bigM + ui;
      lse_acc(m0, s0, x); lse_acc(m1, s1, x);
      lse_acc(m2, s2, x); lse_acc(m3, s3, x);
    }
  }
  unsigned p = (unsigned)((ch*BB + b)*NN) + (unsigned)j0;
  *(float4*)(vpm + p) = make_float4(m0, m1, m2, m3);
  *(float4*)(vps + p) = make_float4(s0, s1, s2, s3);
}

// ---- kernel 7: v-update stage B: combine partials (float4) --------------
__global__ void ot_v_combine(const float* __restrict__ vpm,
                             const float* __restrict__ vps,
                             float* __restrict__ v)
{
  int g = (blockIdx.x*256 + threadIdx.x) * 4;   // BB*NN/1024 = 16 blocks
  int b = g >> 10, j = g & 1023;
  float m0 = -INFINITY, m1 = -INFINITY, m2 = -INFINITY, m3 = -INFINITY;
  float s0 = 0.0f, s1 = 0.0f, s2 = 0.0f, s3 = 0.0f;
  for (int ch = 0; ch < VCH; ++ch) {
    unsigned p = (unsigned)((ch*BB + b)*NN) + (unsigned)j;
    float4 mm = *(const float4*)(vpm + p);
    float4 ss = *(const float4*)(vps + p);
    lse_merge(m0, s0, mm.x, ss.x);
    lse_merge(m1, s1, mm.y, ss.y);
    lse_merge(m2, s2, mm.z, ss.z);
    lse_merge(m3, s3, mm.w, ss.w);
  }
  float4 res = make_float4(LOGB - (m0 + __logf(s0)),
                           LOGB - (m1 + __logf(s1)),
                           LOGB - (m2 + __logf(s2)),
                           LOGB - (m3 + __logf(s3)));
  *(float4*)(v + g) = res;
}

// ---- kernel 8: u-update: row LSE, one block per (b,i) row (b128) --------
__global__ void ot_u(const float* __restrict__ loss,
                     const float* __restrict__ v,
                     const float* __restrict__ loga,
                     const float* __restrict__ validrow,
                     const unsigned int* __restrict__ scal,
                     float* __restrict__ u)
{
  const float* sf = (const float*)scal;
  float scale = sf[5], bigM = sf[6];
  int row = blockIdx.x;                 // BB*NRR = 16400 rows
  int b = row / NRR, i = row % NRR;
  int tid = threadIdx.x;
  bool vr = (i == NN) || (validrow[b*NN + i] != 0.0f);
  float m = -INFINITY, s = 0.0f;
  const float* lrow = loss + (unsigned)(b*NRR + i) * NN;
  const float* vb   = v + b*NN;
  int j0 = tid * 4;                     // 256 threads * 4 = 1024 columns
  float4 vv = *(const float4*)(vb + j0);
  if (vr) {
    float4 lv = *(const float4*)(lrow + j0);
    lse_acc(m, s, scale*lv.x + vv.x);
    lse_acc(m, s, scale*lv.y + vv.y);
    lse_acc(m, s, scale*lv.z + vv.z);
    lse_acc(m, s, scale*lv.w + vv.w);
  } else {
    lse_acc(m, s, bigM + vv.x);
    lse_acc(m, s, bigM + vv.y);
    lse_acc(m, s, bigM + vv.z);
    lse_acc(m, s, bigM + vv.w);
  }
  __shared__ float sm[256], ss[256];
  sm[tid] = m; ss[tid] = s; __syncthreads();
  for (int o = 128; o > 0; o >>= 1) {
    if (tid < o) { float mm = sm[tid], sc = ss[tid];
                   lse_merge(mm, sc, sm[tid+o], ss[tid+o]);
                   sm[tid] = mm; ss[tid] = sc; }
    __syncthreads();
  }
  if (tid == 0) u[row] = loga[row] - (sm[0] + __logf(ss[0]));
}

// ---- kernel 9: final sum(loss * K) partials (float4) --------------------
__global__ void ot_final1(const float* __restrict__ loss,
                          const float* __restrict__ u,
                          const float* __restrict__ v,
                          const float* __restrict__ validrow,
                          const unsigned int* __restrict__ scal,
                          float* __restrict__ part)
{
  const float* sf = (const float*)scal;
  float scale = sf[5], bigM = sf[6];
  float acc = 0.0f;
  int stride = gridDim.x * 256;
  const int TOT4 = (BB * NRR * NN) / 4;     // rows are 1024 floats: no straddle
  for (int t = blockIdx.x*256 + threadIdx.x; t < TOT4; t += stride) {
    int e = t * 4;
    int b = e / (NRR*NN);
    int r = e - b*(NRR*NN);
    int i = r >> 10, j = r & 1023;
    float4 lv = *(const float4*)(loss + (unsigned)e);
    float4 vv = *(const float4*)(v + (unsigned)(b*NN + j));
    float ui = u[b*NRR + i];
    bool vr = (i == NN) || (validrow[b*NN + i] != 0.0f);
    float mx0 = vr ? scale*lv.x : bigM;
    float mx1 = vr ? scale*lv.y : bigM;
    float mx2 = vr ? scale*lv.z : bigM;
    float mx3 = vr ? scale*lv.w : bigM;
    acc += lv.x * __expf(ui + mx0 + vv.x);
    acc += lv.y * __expf(ui + mx1 + vv.y);
    acc += lv.z * __expf(ui + mx2 + vv.z);
    acc += lv.w * __expf(ui + mx3 + vv.w);
  }
  __shared__ float sa[256];
  sa[threadIdx.x] = acc; __syncthreads();
  for (int o = 128; o > 0; o >>= 1) {
    if (threadIdx.x < o) sa[threadIdx.x] += sa[threadIdx.x+o];
    __syncthreads();
  }
  if (threadIdx.x == 0) part[blockIdx.x] = sa[0];
}

// ---- kernel 10: deterministic final reduction ---------------------------
__global__ void ot_final2(const float* __restrict__ part, float* __restrict__ out)
{
  __shared__ float sa[1024];
  int tid = threadIdx.x;
  sa[tid] = part[tid];
  __syncthreads();
  for (int o = 512; o > 0; o >>= 1) { if (tid < o) sa[tid] += sa[tid+o]; __syncthreads(); }
  if (tid == 0) out[0] = sa[0] * (1.0f/(float)BB);
}

// ---- host launcher ------------------------------------------------------
extern "C" void kernel_launch(void* const* d_in, const int* in_sizes, int n_in,
                              void* d_out, int out_size, void* d_ws, size_t ws_size,
                              hipStream_t stream)
{
  (void)in_sizes; (void)n_in; (void)out_size; (void)ws_size;
  const float*         s_logits = (const float*)d_in[0];
  const unsigned char* mask_gt  = (const unsigned char*)d_in[1];
  const float*         xyz      = (const float*)d_in[2];
  const float*         xyz_gt   = (const float*)d_in[3];
  const float*         nph      = (const float*)d_in[4];
  const float*         nph_gt   = (const float*)d_in[5];
  const unsigned char* valid    = (const unsigned char*)d_in[6];
  const float*         log_sig  = (const float*)d_in[7];
  const float*         rnxyz    = (const float*)d_in[8];
  const float*         rnph     = (const float*)d_in[9];
  float* outp = (float*)d_out;

  char* ws = (char*)d_ws;
  size_t off = 0;
  auto take = [&](size_t bytes) -> char* {
    char* p = ws + off;
    off = (off + bytes + 255) & ~(size_t)255;
    return p;
  };
  float* loss     = (float*)take((size_t)BB*NRR*NN*sizeof(float));   // 67.2 MB
  float* gsA      = (float*)take((size_t)BB*NN*4*sizeof(float));
  float* xsB      = (float*)take((size_t)BB*NN*4*sizeof(float));
  float* ri       = (float*)take((size_t)BB*NN*sizeof(float));
  float* cj       = (float*)take((size_t)BB*NN*sizeof(float));
  float* u        = (float*)take((size_t)BB*NRR*sizeof(float));
  float* v        = (float*)take((size_t)BB*NN*sizeof(float));
  float* loga     = (float*)take((size_t)BB*NRR*sizeof(float));
  float* validrow = (float*)take((size_t)BB*NN*sizeof(float));
  float* vpm      = (float*)take((size_t)VCH*BB*NN*sizeof(float));   // 2 MB
  float* vps      = (float*)take((size_t)VCH*BB*NN*sizeof(float));   // 2 MB
  float* part     = (float*)take(1024*sizeof(float));
  unsigned int* hist = (unsigned int*)take(HBINS*sizeof(unsigned int));
  unsigned int* scal = (unsigned int*)take(64*sizeof(unsigned int));

  ot_prep<<<BB, 256, 0, stream>>>(s_logits, mask_gt, xyz, xyz_gt, nph, nph_gt,
                                  valid, log_sig, rnxyz, rnph, gsA, xsB, ri, cj,
                                  loss, loga, validrow, u, v, hist, scal);
  ot_cost_wmma<<<(BB*64*16)/8, 256, 0, stream>>>(gsA, xsB, ri, cj, loss, scal);
  ot_range_init<<<1, 1, 0, stream>>>(scal);
  ot_hist<<<2048, 256, 0, stream>>>(loss, scal, hist);
  ot_select<<<1, 256, 0, stream>>>(hist, scal, 0);
  ot_hist<<<2048, 256, 0, stream>>>(loss, scal, hist);
  ot_select<<<1, 256, 0, stream>>>(hist, scal, 1);
  for (int it = 0; it < NITERS; ++it) {
    ot_v_partial<<<BB*VCH, 256, 0, stream>>>(loss, u, validrow, scal, vpm, vps);
    ot_v_combine<<<(BB*NN)/1024, 256, 0, stream>>>(vpm, vps, v);
    ot_u<<<BB*NRR, 256, 0, stream>>>(loss, v, loga, validrow, scal, u);
  }
  ot_final1<<<1024, 256, 0, stream>>>(loss, u, v, validrow, scal, part);
  ot_final2<<<1, 1024, 0, stream>>>(part, outp);
}